// Propagator2D_51582557225587
// MI455X (gfx1250) — compile-verified
//
#include <hip/hip_runtime.h>

typedef float vf4 __attribute__((ext_vector_type(4)));

// ---------------- static problem config (matches reference) ----------------
namespace {
constexpr int kNZ   = 300;
constexpr int kNX   = 500;
constexpr int kNPAD = 40;
constexpr int kNZP  = kNZ + kNPAD;          // 340
constexpr int kNXP  = kNX + 2 * kNPAD;      // 580
constexpr int kC    = 8;                    // shots
constexpr int kNIMG = 2 * kC;               // B*C = 16
constexpr int kIMG  = kNZP * kNXP;          // 197200
constexpr int kASZ  = kNIMG * kIMG;         // 3155200 elems per output array

constexpr float kDT = 1e-3f;
constexpr float kA1 = (1.0f / 24.0f) / 10.0f;   // a / dx
constexpr float kB1 = (9.0f / 8.0f) / 10.0f;    // b / dx
// dt * (-ln(1e-5) * 3 * 4000 / (2*(dx*npad)^2))
constexpr float kDTCOEF = 4.3173470e-4f;

// tiling: 64 (x) by 16 (z) outputs per 256-thread block; 4-wide per thread
constexpr int TX = 64, TZ = 16;
constexpr int HX = 4;                 // staged x halo (stencil needs 2; 4 keeps 16B alignment)
constexpr int TXH = TX + 2 * HX;      // 72
constexpr int TZH = TZ + 4;           // 20
constexpr int HTILE = TXH * TZH;      // 1440
constexpr int NCHUNK = HTILE / 4;     // 360 16-byte chunks
constexpr int NTHREADS = 256;
} // namespace

// select from the 12-wide x-window (constant-folds after unroll)
#define EW(a, b, c, i) ((i) < 4 ? (a)[(i)] : ((i) < 8 ? (b)[(i) - 4] : (c)[(i) - 8]))

// ---------------- PML damping (analytic, overwrite semantics) ----------------
__device__ __forceinline__ float dax_of(int x) {
  int n;
  if (x < kNPAD)              n = kNPAD - x - 1;
  else if (x >= kNX + kNPAD)  n = x - (kNX + kNPAD);
  else return 1.0f - kDT;     // interior: ax == 1
  float f = (float)n;
  return 1.0f - kDTCOEF * f * f;
}
__device__ __forceinline__ float daz_of(int z) {
  if (z < kNZ) return 1.0f - kDT;   // free surface: top untouched
  float f = (float)(z - kNZ);
  return 1.0f - kDTCOEF * f * f;
}

// ---------------- CDNA5 async global->LDS staging ----------------
__device__ __forceinline__ void wait_async0() {
#if __has_builtin(__builtin_amdgcn_s_wait_asynccnt)
  __builtin_amdgcn_s_wait_asynccnt(0);
#else
  asm volatile("s_wait_asynccnt 0" ::: "memory");
#endif
}

// one 16B chunk: copies 16 bytes from global address sbase + byte_off to the LDS address in lds_off, async (ASYNCcnt)
__device__ __forceinline__ void async_lds_b128(unsigned lds_off, unsigned byte_off,
                                               const float* sbase) {
  asm volatile("global_load_async_to_lds_b128 %0, %1, %2"
               :: "v"(lds_off), "v"(byte_off), "s"(sbase)
               : "memory");
}

// Stage a (TZH x TXH) halo tile of one image into LDS.
// Covers rows [z0-2, z0+TZ+2), cols [x0-HX, x0+TX+HX). 16B chunks never
// straddle the image x borders (0 and 580 are multiples of 4; row stride
// 580*4B is 16B aligned), so each chunk is fully in-image or fully OOB.
__device__ __forceinline__ void stage_halo_tile(const float* __restrict__ img_base,
                                                float* __restrict__ lds_tile,
                                                int z0, int x0, int tid) {
  const vf4 vzero = {0.f, 0.f, 0.f, 0.f};
  for (int ch = tid; ch < NCHUNK; ch += NTHREADS) {
    const int r    = ch / (TXH / 4);
    const int cc   = ch - r * (TXH / 4);
    const int gz   = z0 - 2 + r;
    const int gx0  = x0 - HX + cc * 4;
    const int idx4 = r * TXH + cc * 4;
    if (gz >= 0 && gz < kNZP && gx0 >= 0 && gx0 < kNXP) {
      unsigned lds_off  = (unsigned)(unsigned long long)(lds_tile + idx4);
      unsigned byte_off = (unsigned)((gz * kNXP + gx0) * 4);
      async_lds_b128(lds_off, byte_off, img_base);
    } else {
      *(vf4*)(lds_tile + idx4) = vzero;
    }
  }
}

// ---------------- phase 1: velocity update ----------------
__global__ __launch_bounds__(NTHREADS)
void elastic_vel_kernel(const float* __restrict__ txx, const float* __restrict__ tzz,
                        const float* __restrict__ txz,
                        const float* __restrict__ vx_x, const float* __restrict__ vx_z,
                        const float* __restrict__ vz_x, const float* __restrict__ vz_z,
                        const float* __restrict__ rho,
                        const float* __restrict__ fs, const int* __restrict__ zs,
                        const int* __restrict__ xs, float* __restrict__ out) {
  __shared__ __align__(16) float s_txx[HTILE];
  __shared__ __align__(16) float s_txz[HTILE];
  __shared__ __align__(16) float s_tzz[HTILE];

  const int tid = threadIdx.x;
  const int x0  = blockIdx.x * TX;
  const int z0  = blockIdx.y * TZ;
  const int img = blockIdx.z;
  const int b   = img / kC;
  const int c   = img - b * kC;
  const int imgoff = img * kIMG;
  const int boff   = b * kIMG;

  stage_halo_tile(txx + imgoff, s_txx, z0, x0, tid);
  stage_halo_tile(txz + imgoff, s_txz, z0, x0, tid);
  stage_halo_tile(tzz + imgoff, s_tzz, z0, x0, tid);
  wait_async0();
  __syncthreads();

  // source injection into the staged tzz tile (feeds the conv only)
  if (tid == 0) {
    const int sz = zs[c];
    const int sx = xs[c] + kNPAD;
    const int lz = sz - (z0 - 2);
    const int lx = sx - (x0 - HX);
    if (lz >= 0 && lz < TZH && lx >= 0 && lx < TXH)
      s_tzz[lz * TXH + lx] += fs[b];
  }
  __syncthreads();

  const int lane = tid & 15;        // 16 lanes of 4 floats cover 64 columns
  const int row  = tid >> 4;        // 16 rows
  const int cx   = lane * 4;
  const int x    = x0 + cx;
  const int z    = z0 + row;
  if (x < kNXP && z < kNZP) {
    const int lb = (row + 2) * TXH + (cx + HX);    // 16B aligned (cx,HX mult of 4)

    // x-windows: e[-4..7] around the 4 outputs, three aligned b128 LDS reads
    const vf4 xxa = *(const vf4*)(s_txx + lb - 4);
    const vf4 xxb = *(const vf4*)(s_txx + lb);
    const vf4 xxc = *(const vf4*)(s_txx + lb + 4);
    const vf4 xza = *(const vf4*)(s_txz + lb - 4);
    const vf4 xzb = *(const vf4*)(s_txz + lb);
    const vf4 xzc = *(const vf4*)(s_txz + lb + 4);
    // z-taps (aligned b128 rows)
    const vf4 xzm2 = *(const vf4*)(s_txz + lb - 2 * TXH);
    const vf4 xzm1 = *(const vf4*)(s_txz + lb - TXH);
    const vf4 xzp1 = *(const vf4*)(s_txz + lb + TXH);
    const vf4 zzm1 = *(const vf4*)(s_tzz + lb - TXH);
    const vf4 zz0  = *(const vf4*)(s_tzz + lb);
    const vf4 zzp1 = *(const vf4*)(s_tzz + lb + TXH);
    const vf4 zzp2 = *(const vf4*)(s_tzz + lb + 2 * TXH);

    vf4 v_txx_x, v_txz_x, v_txz_z, v_tzz_z, dax4;
#pragma unroll
    for (int j = 0; j < 4; ++j) {
      // KX(txx): taps at -1,0,+1,+2 ; KXU(txz): -2,-1,0,+1 (center index 4+j)
      v_txx_x[j] = kA1 * EW(xxa, xxb, xxc, 3 + j) - kB1 * EW(xxa, xxb, xxc, 4 + j)
                 + kB1 * EW(xxa, xxb, xxc, 5 + j) - kA1 * EW(xxa, xxb, xxc, 6 + j);
      v_txz_x[j] = kA1 * EW(xza, xzb, xzc, 2 + j) - kB1 * EW(xza, xzb, xzc, 3 + j)
                 + kB1 * EW(xza, xzb, xzc, 4 + j) - kA1 * EW(xza, xzb, xzc, 5 + j);
      // KZU(txz): z taps -2,-1,0,+1 ; KZ(tzz): -1,0,+1,+2
      v_txz_z[j] = kA1 * xzm2[j] - kB1 * xzm1[j] + kB1 * xzb[j] - kA1 * xzp1[j];
      v_tzz_z[j] = kA1 * zzm1[j] - kB1 * zz0[j] + kB1 * zzp1[j] - kA1 * zzp2[j];
      dax4[j]    = dax_of(x + j);
    }

    const int off = imgoff + z * kNXP + x;          // multiple of 4 -> 16B aligned
    const vf4 r4  = *(const vf4*)(rho + boff + z * kNXP + x);
    const float daz = daz_of(z);

    const vf4 ovxx = __builtin_nontemporal_load((const vf4*)(vx_x + off));
    const vf4 ovxz = __builtin_nontemporal_load((const vf4*)(vx_z + off));
    const vf4 ovzx = __builtin_nontemporal_load((const vf4*)(vz_x + off));
    const vf4 ovzz = __builtin_nontemporal_load((const vf4*)(vz_z + off));

    vf4 nvx_x, nvx_z, nvz_x, nvz_z;
#pragma unroll
    for (int j = 0; j < 4; ++j) {
      const float rinv = 1.0f / r4[j];
      nvx_x[j] = dax4[j] * ovxx[j] + kDT * v_txx_x[j] * rinv;
      nvx_z[j] = daz     * ovxz[j] + kDT * v_txz_z[j] * rinv;
      nvz_x[j] = dax4[j] * ovzx[j] + kDT * v_txz_x[j] * rinv;
      nvz_z[j] = daz     * ovzz[j] + kDT * v_tzz_z[j] * rinv;
    }

    // vx, vz are re-read by the stress kernel: keep them L2-resident
    *(vf4*)(out + 0 * kASZ + off) = nvx_x + nvx_z;
    *(vf4*)(out + 1 * kASZ + off) = nvz_x + nvz_z;
    // split states are write-once: stream past the caches
    __builtin_nontemporal_store(nvx_x, (vf4*)(out + 5 * kASZ + off));
    __builtin_nontemporal_store(nvx_z, (vf4*)(out + 6 * kASZ + off));
    __builtin_nontemporal_store(nvz_x, (vf4*)(out + 7 * kASZ + off));
    __builtin_nontemporal_store(nvz_z, (vf4*)(out + 8 * kASZ + off));
  }
}

// ---------------- phase 2: stress update ----------------
__global__ __launch_bounds__(NTHREADS)
void elastic_stress_kernel(const float* __restrict__ txx_x_in, const float* __restrict__ txx_z_in,
                           const float* __restrict__ tzz_x_in, const float* __restrict__ tzz_z_in,
                           const float* __restrict__ txz_x_in, const float* __restrict__ txz_z_in,
                           const float* __restrict__ vp, const float* __restrict__ vsq,
                           const float* __restrict__ rho, float* __restrict__ out) {
  __shared__ __align__(16) float s_vx[HTILE];
  __shared__ __align__(16) float s_vz[HTILE];

  const int tid = threadIdx.x;
  const int x0  = blockIdx.x * TX;
  const int z0  = blockIdx.y * TZ;
  const int img = blockIdx.z;
  const int b   = img / kC;
  const int imgoff = img * kIMG;
  const int boff   = b * kIMG;

  // vx, vz were just written to d_out by the velocity kernel (L2-hot)
  stage_halo_tile(out + 0 * kASZ + imgoff, s_vx, z0, x0, tid);
  stage_halo_tile(out + 1 * kASZ + imgoff, s_vz, z0, x0, tid);
  wait_async0();
  __syncthreads();

  const int lane = tid & 15;
  const int row  = tid >> 4;
  const int cx   = lane * 4;
  const int x    = x0 + cx;
  const int z    = z0 + row;
  if (x < kNXP && z < kNZP) {
    const int lb = (row + 2) * TXH + (cx + HX);

    const vf4 vxa = *(const vf4*)(s_vx + lb - 4);
    const vf4 vxb = *(const vf4*)(s_vx + lb);
    const vf4 vxc = *(const vf4*)(s_vx + lb + 4);
    const vf4 vza = *(const vf4*)(s_vz + lb - 4);
    const vf4 vzb = *(const vf4*)(s_vz + lb);
    const vf4 vzc = *(const vf4*)(s_vz + lb + 4);
    const vf4 vxm1 = *(const vf4*)(s_vx + lb - TXH);
    const vf4 vxp1 = *(const vf4*)(s_vx + lb + TXH);
    const vf4 vxp2 = *(const vf4*)(s_vx + lb + 2 * TXH);
    const vf4 vzm2 = *(const vf4*)(s_vz + lb - 2 * TXH);
    const vf4 vzm1 = *(const vf4*)(s_vz + lb - TXH);
    const vf4 vzp1 = *(const vf4*)(s_vz + lb + TXH);

    vf4 v_vx_x, v_vx_z, v_vz_x, v_vz_z, dax4;
#pragma unroll
    for (int j = 0; j < 4; ++j) {
      // KXU(vx): -2,-1,0,+1 ; KX(vz): -1,0,+1,+2 (center 4+j)
      v_vx_x[j] = kA1 * EW(vxa, vxb, vxc, 2 + j) - kB1 * EW(vxa, vxb, vxc, 3 + j)
                + kB1 * EW(vxa, vxb, vxc, 4 + j) - kA1 * EW(vxa, vxb, vxc, 5 + j);
      v_vz_x[j] = kA1 * EW(vza, vzb, vzc, 3 + j) - kB1 * EW(vza, vzb, vzc, 4 + j)
                + kB1 * EW(vza, vzb, vzc, 5 + j) - kA1 * EW(vza, vzb, vzc, 6 + j);
      // KZ(vx): -1,0,+1,+2 ; KZU(vz): -2,-1,0,+1
      v_vx_z[j] = kA1 * vxm1[j] - kB1 * vxb[j] + kB1 * vxp1[j] - kA1 * vxp2[j];
      v_vz_z[j] = kA1 * vzm2[j] - kB1 * vzm1[j] + kB1 * vzb[j] - kA1 * vzp1[j];
      dax4[j]   = dax_of(x + j);
    }

    const int off  = imgoff + z * kNXP + x;
    const int offm = boff + z * kNXP + x;
    const vf4 r4 = *(const vf4*)(rho + offm);
    const vf4 p4 = *(const vf4*)(vp + offm);
    const vf4 s4 = *(const vf4*)(vsq + offm);
    const float daz = daz_of(z);

    const vf4 otxxx = __builtin_nontemporal_load((const vf4*)(txx_x_in + off));
    const vf4 otxxz = __builtin_nontemporal_load((const vf4*)(txx_z_in + off));
    const vf4 otzzx = __builtin_nontemporal_load((const vf4*)(tzz_x_in + off));
    const vf4 otzzz = __builtin_nontemporal_load((const vf4*)(tzz_z_in + off));
    const vf4 otxzx = __builtin_nontemporal_load((const vf4*)(txz_x_in + off));
    const vf4 otxzz = __builtin_nontemporal_load((const vf4*)(txz_z_in + off));

    vf4 ntxx_x, ntxx_z, ntzz_x, ntzz_z, ntxz_x, ntxz_z;
#pragma unroll
    for (int j = 0; j < 4; ++j) {
      const float lam2mu = p4[j] * p4[j] * r4[j];
      const float mu     = s4[j] * s4[j] * r4[j];
      const float lam    = lam2mu - 2.0f * mu;
      ntxx_x[j] = dax4[j] * otxxx[j] + kDT * lam2mu * v_vx_x[j];
      ntxx_z[j] = daz     * otxxz[j] + kDT * lam    * v_vz_z[j];
      ntzz_x[j] = dax4[j] * otzzx[j] + kDT * lam    * v_vx_x[j];
      ntzz_z[j] = daz     * otzzz[j] + kDT * lam2mu * v_vz_z[j];
      ntxz_x[j] = dax4[j] * otxzx[j] + kDT * mu     * v_vz_x[j];
      ntxz_z[j] = daz     * otxzz[j] + kDT * mu     * v_vx_z[j];
    }

    __builtin_nontemporal_store(ntxx_x + ntxx_z, (vf4*)(out +  2 * kASZ + off));  // txx
    __builtin_nontemporal_store(ntzz_x + ntzz_z, (vf4*)(out +  3 * kASZ + off));  // tzz
    __builtin_nontemporal_store(ntxz_x + ntxz_z, (vf4*)(out +  4 * kASZ + off));  // txz
    __builtin_nontemporal_store(ntxx_x, (vf4*)(out +  9 * kASZ + off));
    __builtin_nontemporal_store(ntxx_z, (vf4*)(out + 10 * kASZ + off));
    __builtin_nontemporal_store(ntzz_x, (vf4*)(out + 11 * kASZ + off));
    __builtin_nontemporal_store(ntzz_z, (vf4*)(out + 12 * kASZ + off));
    __builtin_nontemporal_store(ntxz_x, (vf4*)(out + 13 * kASZ + off));
    __builtin_nontemporal_store(ntxz_z, (vf4*)(out + 14 * kASZ + off));
  }
}

// ---------------- host side ----------------
extern "C" void kernel_launch(void* const* d_in, const int* in_sizes, int n_in,
                              void* d_out, int out_size, void* d_ws, size_t ws_size,
                              hipStream_t stream) {
  (void)in_sizes; (void)n_in; (void)out_size; (void)d_ws; (void)ws_size;

  const float* txx   = (const float*)d_in[0];
  const float* tzz   = (const float*)d_in[1];
  const float* txz   = (const float*)d_in[2];
  const float* vx_x  = (const float*)d_in[3];
  const float* vx_z  = (const float*)d_in[4];
  const float* vz_x  = (const float*)d_in[5];
  const float* vz_z  = (const float*)d_in[6];
  const float* txx_x = (const float*)d_in[7];
  const float* txx_z = (const float*)d_in[8];
  const float* tzz_x = (const float*)d_in[9];
  const float* tzz_z = (const float*)d_in[10];
  const float* txz_x = (const float*)d_in[11];
  const float* txz_z = (const float*)d_in[12];
  const float* vp    = (const float*)d_in[13];
  const float* vs    = (const float*)d_in[14];
  const float* rho   = (const float*)d_in[15];
  const float* fs    = (const float*)d_in[16];
  const int*   zs    = (const int*)d_in[17];
  const int*   xs    = (const int*)d_in[18];
  float* out = (float*)d_out;

  dim3 grid((kNXP + TX - 1) / TX, (kNZP + TZ - 1) / TZ, kNIMG);  // (10, 22, 16)
  dim3 block(NTHREADS);

  elastic_vel_kernel<<<grid, block, 0, stream>>>(txx, tzz, txz, vx_x, vx_z, vz_x, vz_z,
                                                 rho, fs, zs, xs, out);
  elastic_stress_kernel<<<grid, block, 0, stream>>>(txx_x, txx_z, tzz_x, tzz_z, txz_x, txz_z,
                                                    vp, vs, rho, out);
}